// MDTracker_70978629533714
// MI455X (gfx1250) — compile-verified
//
#include <hip/hip_runtime.h>
#include <hip/hip_bf16.h>
#include <cstddef>

typedef float v2f __attribute__((ext_vector_type(2)));
typedef float v8f __attribute__((ext_vector_type(8)));

// Device-pass-only existence check (host pass doesn't know amdgcn builtins).
#if defined(__HIP_DEVICE_COMPILE__)
#if !__has_builtin(__builtin_amdgcn_wmma_f32_16x16x4_f32)
#error "DEVICE: wmma_f32_16x16x4_f32 builtin not available"
#endif
#endif

__device__ inline v8f wmma_16x16x4_f32(v2f a, v2f b, v8f c) {
#if defined(__HIP_DEVICE_COMPILE__)
    return __builtin_amdgcn_wmma_f32_16x16x4_f32(false, a, false, b, (short)0, c, false, false);
#else
    (void)a; (void)b;
    return c;   // host stub, never executed
#endif
}

// ---------------- problem constants ----------------
constexpr int S_     = 8;
constexpr int NTRK   = 1024;
constexpr int CLAT   = 128;
constexpr int H8_    = 48;
constexpr int W8_    = 64;
constexpr int EMB_   = 456;
constexpr int HID_   = 384;
constexpr int NH_    = 8;
constexpr int HD_    = 48;
constexpr int PP_    = 49;          // (2*RADIUS+1)^2
constexpr int MTOK   = NTRK * S_;   // 8192
constexpr int OUTSTEP = S_ * NTRK * 3; // 24576 floats per prediction slab

__device__ inline int iclamp(int v, int lo, int hi) { return v < lo ? lo : (v > hi ? hi : v); }

__device__ inline float gelu_tanh(float v) {
    const float c = 0.7978845608028654f;   // sqrt(2/pi)
    float u = c * (v + 0.044715f * v * v * v);
    return 0.5f * v * (1.f + tanhf(u));
}

// ================= WMMA f32 GEMM: C = act(A@B + bias [+ C]) =================
// MODE 0: store (A@B + bias)
// MODE 1: store C + (A@B + bias)      (residual, in-place on C)
// MODE 2: store gelu(A@B + bias)
// EDGE  : Nc not a multiple of 64 -> clamp addresses + select-zero (branchless)
// Tile: 32 rows x 64 cols per wave (2 M-subtiles share 4 B fragments).
template<int MODE, bool EDGE>
__global__ __launch_bounds__(256)
void gemm_wmma_f32(const float* __restrict__ A, const float* __restrict__ B,
                   const float* __restrict__ bias, float* C,
                   int M, int K, int Nc)
{
    const int lane = threadIdx.x & 31;
    const int wv   = threadIdx.x >> 5;
    const int tilesN = (Nc + 63) >> 6;
    const int tilesM = M >> 5;                 // 32 rows per wave
    const int wid = blockIdx.x * 8 + wv;
    const int tm = wid / tilesN;
    const int tn = wid - tm * tilesN;
    if (tm >= tilesM) return;

    const int row0 = tm << 5;
    const int col0 = tn << 6;
    const int l15  = lane & 15;
    const int half = lane >> 4;                // 0: K=0,1   1: K=2,3  (ISA 16x4 f32 A layout)
    const int kk   = half << 1;

    const float* Arow0 = A + (size_t)(row0 + l15) * K + kk;
    const float* Arow1 = Arow0 + (size_t)16 * K;

    int  c[4];
    int  ccl[4];
    bool ok[4];
#pragma unroll
    for (int j = 0; j < 4; ++j) {
        c[j]  = col0 + 16 * j + l15;
        ok[j] = !EDGE || (c[j] < Nc);
        ccl[j] = EDGE ? (c[j] < Nc ? c[j] : Nc - 1) : c[j];
    }

    v8f acc[2][4];
#pragma unroll
    for (int g = 0; g < 2; ++g)
#pragma unroll
        for (int j = 0; j < 4; ++j) acc[g][j] = (v8f){};

    for (int k0 = 0; k0 < K; k0 += 4) {
        v2f a0 = *reinterpret_cast<const v2f*>(Arow0 + k0);   // 8B aligned: K,kk even
        v2f a1 = *reinterpret_cast<const v2f*>(Arow1 + k0);
        const float* Br0 = B + (size_t)(k0 + kk) * Nc;
        const float* Br1 = Br0 + Nc;
        v2f b[4];
#pragma unroll
        for (int j = 0; j < 4; ++j) {
            float x0 = Br0[ccl[j]];          // unconditional (clamped) loads
            float x1 = Br1[ccl[j]];
            b[j].x = ok[j] ? x0 : 0.f;       // v_cndmask, no exec-mask branches
            b[j].y = ok[j] ? x1 : 0.f;
        }
#pragma unroll
        for (int j = 0; j < 4; ++j) {
            acc[0][j] = wmma_16x16x4_f32(a0, b[j], acc[0][j]);
            acc[1][j] = wmma_16x16x4_f32(a1, b[j], acc[1][j]);
        }
        __builtin_prefetch(Br0 + 4 * (size_t)Nc + ccl[0], 0, 1);  // global_prefetch_b8 for next k-group
    }

#pragma unroll
    for (int j = 0; j < 4; ++j) {
        if (EDGE && c[j] >= Nc) continue;
        float bv = bias[c[j]];
#pragma unroll
        for (int g = 0; g < 2; ++g) {
#pragma unroll
            for (int i = 0; i < 8; ++i) {
                // C layout: VGPR i -> M=i (lanes 0-15), M=i+8 (lanes 16-31)
                int r = row0 + (g << 4) + i + (half << 3);
                float v = acc[g][j][i] + bv;
                size_t idx = (size_t)r * Nc + c[j];
                if (MODE == 1)      v += C[idx];
                else if (MODE == 2) v = gelu_tanh(v);
                C[idx] = v;
            }
        }
    }
}

static inline void launch_gemm(const float* A, const float* B, const float* bias, float* C,
                               int M, int K, int Nc, int mode, hipStream_t st)
{
    int tilesN = (Nc + 63) / 64;
    int tilesM = M / 32;
    int blocks = (tilesM * tilesN + 7) / 8;
    bool edge = (Nc & 63) != 0;
#define GL(MD, ED) gemm_wmma_f32<MD, ED><<<blocks, 256, 0, st>>>(A, B, bias, C, M, K, Nc)
    if (!edge) {
        if (mode == 0) GL(0, false); else if (mode == 1) GL(1, false); else GL(2, false);
    } else {
        if (mode == 0) GL(0, true);  else if (mode == 1) GL(1, true);  else GL(2, true);
    }
#undef GL
}

// ================= pyramid downsample (2x2 mean) =================
__global__ __launch_bounds__(256)
void pyramid_down(const float* __restrict__ src, float* __restrict__ dst, int Hs, int Ws)
{
    int Hd = Hs >> 1, Wd = Ws >> 1;
    int total = S_ * CLAT * Hd * Wd;
    int i = blockIdx.x * 256 + threadIdx.x;
    if (i >= total) return;
    int x = i % Wd;
    int y = (i / Wd) % Hd;
    int sc = i / (Wd * Hd);
    const float* p = src + (size_t)sc * Hs * Ws;
    int i0 = (2 * y) * Ws + 2 * x;
    dst[i] = 0.25f * (p[i0] + p[i0 + 1] + p[i0 + Ws] + p[i0 + Ws + 1]);
}

// ================= pos / time embeddings (fixed for whole run) =================
__global__ __launch_bounds__(256)
void pos_time_embed(const float* __restrict__ coords_init,
                    float* __restrict__ posE, float* __restrict__ timE)
{
    const float LOG1E4 = 9.210340371976184f;
    int idx = blockIdx.x * 256 + threadIdx.x;
    if (idx < NTRK * EMB_) {
        int d = idx % EMB_, n = idx / EMB_;
        float xc = coords_init[(size_t)n * 3 + 0];   // coords_init[0,0,n,0]
        float yc = coords_init[(size_t)n * 3 + 1];
        int x0 = iclamp((int)floorf(xc), 0, W8_ - 2);
        int y0 = iclamp((int)floorf(yc), 0, H8_ - 2);
        float wx = xc - (float)x0, wy = yc - (float)y0;
        auto pm = [&](int yy, int xx) -> float {
            int dd = d; float pos;
            if (dd < 228) { pos = (float)yy; }
            else          { pos = (float)xx; dd -= 228; }
            int j = dd % 114;
            bool iscos = dd >= 114;
            float om = __expf(-(float)j * (LOG1E4 / 114.f));
            float a = pos * om;
            return iscos ? cosf(a) : sinf(a);
        };
        float v = pm(y0, x0)     * (1.f - wy) * (1.f - wx)
                + pm(y0, x0 + 1) * (1.f - wy) * wx
                + pm(y0 + 1, x0) * wy * (1.f - wx)
                + pm(y0 + 1, x0 + 1) * wy * wx;
        posE[idx] = v;
    } else if (idx < NTRK * EMB_ + S_ * EMB_) {
        int r = idx - NTRK * EMB_;
        int d = r % EMB_, s = r / EMB_;
        int j = d % 228;
        bool iscos = d >= 228;
        float om = __expf(-(float)j * (LOG1E4 / 228.f));
        float a = (float)s * om;
        timE[r] = iscos ? cosf(a) : sinf(a);
    }
}

// ================= state init: coords copy, ffeats (S,N,C)->(N,S,C) =================
__global__ __launch_bounds__(256)
void init_state(const float* __restrict__ coords_init, const float* __restrict__ feat_init,
                float* __restrict__ coords, float* __restrict__ ffeats)
{
    int idx = blockIdx.x * 256 + threadIdx.x;
    if (idx < S_ * NTRK * 3) coords[idx] = coords_init[idx];
    if (idx < S_ * NTRK * CLAT) {
        int c = idx & (CLAT - 1);
        int t = idx >> 7;            // t = s*NTRK + n  (feat_init order)
        int s = t / NTRK, n = t % NTRK;
        ffeats[((size_t)n * S_ + s) * CLAT + c] = feat_init[idx];
    }
}

// ================= fused correlation + bilinear sampling =================
__device__ inline void corr_level(const float* __restrict__ fm, int Hl, int Wl,
                                  float cx, float cy,
                                  float t0, float t1, float t2, float t3,
                                  int lane, float* __restrict__ outp)
{
    const size_t HW = (size_t)Hl * Wl;
    const float* p0 = fm + (size_t)lane * HW;
    const float* p1 = p0 + 32 * HW;
    const float* p2 = p0 + 64 * HW;
    const float* p3 = p0 + 96 * HW;
    const float scale = 0.0883883476483184f;   // 1/sqrt(128)
    for (int j = 0; j < PP_; ++j) {
        float xs = cx + (float)(j % 7) - 3.f;
        float ys = cy + (float)(j / 7) - 3.f;
        int x0 = iclamp((int)floorf(xs), 0, Wl - 2);
        int y0 = iclamp((int)floorf(ys), 0, Hl - 2);
        float wx = xs - (float)x0, wy = ys - (float)y0;
        float w00 = (1.f - wy) * (1.f - wx), w01 = (1.f - wy) * wx;
        float w10 = wy * (1.f - wx),         w11 = wy * wx;
        int i00 = y0 * Wl + x0;
        float acc;
        acc  = t0 * (w00 * p0[i00] + w01 * p0[i00 + 1] + w10 * p0[i00 + Wl] + w11 * p0[i00 + Wl + 1]);
        acc += t1 * (w00 * p1[i00] + w01 * p1[i00 + 1] + w10 * p1[i00 + Wl] + w11 * p1[i00 + Wl + 1]);
        acc += t2 * (w00 * p2[i00] + w01 * p2[i00 + 1] + w10 * p2[i00 + Wl] + w11 * p2[i00 + Wl + 1]);
        acc += t3 * (w00 * p3[i00] + w01 * p3[i00 + 1] + w10 * p3[i00 + Wl] + w11 * p3[i00 + Wl + 1]);
        for (int off = 16; off > 0; off >>= 1) acc += __shfl_xor(acc, off, 32);
        if (lane == 0) outp[j] = acc * scale;
    }
}

__global__ __launch_bounds__(256)
void corr_sample(const float* __restrict__ fm0, const float* __restrict__ fm1,
                 const float* __restrict__ fm2, const float* __restrict__ fm3,
                 const float* __restrict__ ffeats, const float* __restrict__ coords,
                 float* __restrict__ x)
{
    int wave = blockIdx.x * 8 + (threadIdx.x >> 5);
    int lane = threadIdx.x & 31;
    if (wave >= S_ * NTRK) return;
    int s = wave / NTRK, n = wave % NTRK;
    const float* tf = ffeats + ((size_t)n * S_ + s) * CLAT;
    float t0 = tf[lane], t1 = tf[lane + 32], t2 = tf[lane + 64], t3 = tf[lane + 96];
    float cx = coords[((size_t)s * NTRK + n) * 3 + 0];
    float cy = coords[((size_t)s * NTRK + n) * 3 + 1];
    float* outp = x + ((size_t)n * S_ + s) * EMB_ + 130;
    corr_level(fm0 + (size_t)s * CLAT * 48 * 64, 48, 64, cx,          cy,          t0, t1, t2, t3, lane, outp);
    corr_level(fm1 + (size_t)s * CLAT * 24 * 32, 24, 32, cx * 0.5f,   cy * 0.5f,   t0, t1, t2, t3, lane, outp + PP_);
    corr_level(fm2 + (size_t)s * CLAT * 12 * 16, 12, 16, cx * 0.25f,  cy * 0.25f,  t0, t1, t2, t3, lane, outp + 2 * PP_);
    corr_level(fm3 + (size_t)s * CLAT *  6 *  8,  6,  8, cx * 0.125f, cy * 0.125f, t0, t1, t2, t3, lane, outp + 3 * PP_);
}

// ================= flow sincos embedding + small (195x130) GEMM =================
__global__ __launch_bounds__(256)
void flow_embed_gemm(const float* __restrict__ coords, const float* __restrict__ Wf,
                     const float* __restrict__ bf, float* __restrict__ x)
{
    __shared__ float e[195];
    int t = blockIdx.x;            // token = n*S + s
    int n = t / S_, s = t % S_;
    int tid = threadIdx.x;
    float f0 = coords[((size_t)s * NTRK + n) * 3 + 0] - coords[(size_t)n * 3 + 0];
    float f1 = coords[((size_t)s * NTRK + n) * 3 + 1] - coords[(size_t)n * 3 + 1];
    float f2 = coords[((size_t)s * NTRK + n) * 3 + 2] - coords[(size_t)n * 3 + 2];
    if (tid < 192) {
        int comp = tid / 64;
        int r = tid % 64;
        int j = r >> 1;
        float v = (comp == 0) ? f0 : (comp == 1 ? f1 : f2);
        float a = v * (float)j * 31.25f;     // div[j] = 2j * 1000/64
        e[tid] = (r & 1) ? cosf(a) : sinf(a);
    } else if (tid == 192) e[192] = f0;
    else if (tid == 193)   e[193] = f1;
    else if (tid == 194)   e[194] = f2;
    __syncthreads();
    if (tid < 130) {
        float acc = bf[tid];
        for (int r = 0; r < 195; ++r) acc += e[r] * Wf[r * 130 + tid];
        x[(size_t)t * EMB_ + tid] = acc;
    }
}

// ================= assemble x: fill [326..456), add pos/time embeds =================
__global__ __launch_bounds__(256)
void assemble_x(float* __restrict__ x, const float* __restrict__ ffeats,
                const float* __restrict__ track_mask, const float* __restrict__ vis_init,
                const float* __restrict__ posE, const float* __restrict__ timE)
{
    int idx = blockIdx.x * 256 + threadIdx.x;
    if (idx >= MTOK * EMB_) return;
    int d = idx % EMB_;
    int t = idx / EMB_;
    int n = t / S_, s = t % S_;
    float base;
    if (d < 326)        base = x[idx];                                  // flow_cat | corr (raw)
    else if (d < 454)   base = ffeats[(size_t)t * CLAT + (d - 326)];
    else if (d == 454)  base = track_mask[(size_t)s * NTRK + n];
    else                base = vis_init[(size_t)s * NTRK + n];
    x[idx] = base + posE[(size_t)n * EMB_ + d] + timE[(size_t)s * EMB_ + d];
}

// ================= LayerNorm (wave per token) =================
__global__ __launch_bounds__(256)
void layernorm(const float* __restrict__ in, const float* __restrict__ w,
               const float* __restrict__ b, float* __restrict__ out,
               int dim, int inStride, int inOff)
{
    int token = blockIdx.x * 8 + (threadIdx.x >> 5);
    int lane = threadIdx.x & 31;
    if (token >= MTOK) return;
    const float* p = in + (size_t)token * inStride + inOff;
    float s = 0.f, sq = 0.f;
    for (int d = lane; d < dim; d += 32) { float v = p[d]; s += v; sq += v * v; }
    for (int off = 16; off > 0; off >>= 1) { s += __shfl_xor(s, off, 32); sq += __shfl_xor(sq, off, 32); }
    float m = s / (float)dim;
    float var = sq / (float)dim - m * m;
    float rs = rsqrtf(var + 1e-5f);
    float* q = out + (size_t)token * dim;
    for (int d = lane; d < dim; d += 32) q[d] = (p[d] - m) * rs * w[d] + b[d];
}

// ================= tiny attention over time (thread per n,head,ts) =================
__global__ __launch_bounds__(256)
void attention(const float* __restrict__ qkv, float* __restrict__ o)
{
    int idx = blockIdx.x * 256 + threadIdx.x;
    if (idx >= NTRK * NH_ * S_) return;
    int ts = idx % S_;
    int head = (idx / S_) % NH_;
    int n = idx / (S_ * NH_);
    const float* base = qkv + (size_t)n * S_ * (3 * HID_);
    float q[HD_];
    const float* qp = base + (size_t)ts * (3 * HID_) + head * HD_;
#pragma unroll
    for (int d = 0; d < HD_; ++d) q[d] = qp[d];
    const float scale = 0.14433756729740643f;  // 48^-0.5
    float lg[S_];
    float mx = -1e30f;
    for (int ss = 0; ss < S_; ++ss) {
        const float* kp = base + (size_t)ss * (3 * HID_) + HID_ + head * HD_;
        float acc = 0.f;
#pragma unroll
        for (int d = 0; d < HD_; ++d) acc += q[d] * kp[d];
        lg[ss] = acc * scale;
        mx = fmaxf(mx, lg[ss]);
    }
    float sum = 0.f;
    for (int ss = 0; ss < S_; ++ss) { lg[ss] = __expf(lg[ss] - mx); sum += lg[ss]; }
    float rs = 1.f / sum;
    float ov[HD_];
#pragma unroll
    for (int d = 0; d < HD_; ++d) ov[d] = 0.f;
    for (int ss = 0; ss < S_; ++ss) {
        const float* vp = base + (size_t)ss * (3 * HID_) + 2 * HID_ + head * HD_;
        float w = lg[ss] * rs;
#pragma unroll
        for (int d = 0; d < HD_; ++d) ov[d] += w * vp[d];
    }
    float* op = o + (size_t)(n * S_ + ts) * HID_ + head * HD_;
#pragma unroll
    for (int d = 0; d < HD_; ++d) op[d] = ov[d];
}

// ================= state update + prediction write =================
__global__ __launch_bounds__(256)
void update_state(const float* __restrict__ delta, const float* __restrict__ upd,
                  float* __restrict__ ffeats, float* __restrict__ coords,
                  float* __restrict__ out_preds)
{
    int idx = blockIdx.x * 256 + threadIdx.x;
    if (idx >= MTOK * CLAT) return;
    int t = idx >> 7;
    int c = idx & (CLAT - 1);
    ffeats[idx] += upd[idx];
    if (c == 0) {
        int n = t / S_, s = t % S_;
        for (int k = 0; k < 3; ++k) {
            size_t ci = ((size_t)s * NTRK + n) * 3 + k;
            float v = coords[ci] + delta[(size_t)t * 131 + k];
            coords[ci] = v;
            out_preds[ci] = (k < 2) ? v * 8.f : v;
        }
    }
}

// ================= visibility head (broadcast to 3 comps) =================
__global__ __launch_bounds__(256)
void vis_head(const float* __restrict__ ffeats, const float* __restrict__ Wv,
              const float* __restrict__ bv, float* __restrict__ out)
{
    int idx = blockIdx.x * 256 + threadIdx.x;
    if (idx >= S_ * NTRK) return;
    int s = idx / NTRK, n = idx % NTRK;
    float acc = bv[0];
    const float* p = ffeats + (size_t)(n * S_ + s) * CLAT;
    for (int c = 0; c < CLAT; ++c) acc += p[c] * Wv[c];
    float* q = out + (size_t)(s * NTRK + n) * 3;
    q[0] = acc; q[1] = acc; q[2] = acc;
}

// =====================================================================
extern "C" void kernel_launch(void* const* d_in, const int* in_sizes, int n_in,
                              void* d_out, int out_size, void* d_ws, size_t ws_size,
                              hipStream_t stream)
{
    const float* fmaps       = (const float*)d_in[0];
    const float* coords_init = (const float*)d_in[1];
    const float* feat_init   = (const float*)d_in[2];
    const float* vis_init    = (const float*)d_in[3];
    const float* track_mask  = (const float*)d_in[4];
    const float* W_flow = (const float*)d_in[5];
    const float* b_flow = (const float*)d_in[6];
    const float* W_in   = (const float*)d_in[7];
    const float* b_in   = (const float*)d_in[8];
    const float* ln1_w  = (const float*)d_in[9];
    const float* ln1_b  = (const float*)d_in[10];
    const float* Wqkv   = (const float*)d_in[11];
    const float* bqkv   = (const float*)d_in[12];
    const float* Wproj  = (const float*)d_in[13];
    const float* bproj  = (const float*)d_in[14];
    const float* ln2_w  = (const float*)d_in[15];
    const float* ln2_b  = (const float*)d_in[16];
    const float* Wm1    = (const float*)d_in[17];
    const float* bm1    = (const float*)d_in[18];
    const float* Wm2    = (const float*)d_in[19];
    const float* bm2    = (const float*)d_in[20];
    const float* W_out  = (const float*)d_in[21];
    const float* b_out  = (const float*)d_in[22];
    const float* gn_w   = (const float*)d_in[23];
    const float* gn_b   = (const float*)d_in[24];
    const float* W_ffu  = (const float*)d_in[25];
    const float* b_ffu  = (const float*)d_in[26];
    const float* W_vis  = (const float*)d_in[27];
    const float* b_vis  = (const float*)d_in[28];
    (void)in_sizes; (void)n_in; (void)ws_size;

    int iters = out_size / OUTSTEP - 1;
    if (iters < 1) iters = 1;

    float* out = (float*)d_out;
    float* ws  = (float*)d_ws;
    size_t off = 0;
    auto alloc = [&](size_t nf) { float* p = ws + off; off += nf; return p; };
    float* pyr1   = alloc((size_t)S_ * CLAT * 24 * 32);
    float* pyr2   = alloc((size_t)S_ * CLAT * 12 * 16);
    float* pyr3   = alloc((size_t)S_ * CLAT * 6 * 8);
    float* posE   = alloc((size_t)NTRK * EMB_);
    float* timE   = alloc((size_t)S_ * EMB_);
    float* coords = alloc((size_t)S_ * NTRK * 3);
    float* ffeats = alloc((size_t)MTOK * CLAT);
    float* xbuf   = alloc((size_t)MTOK * EMB_);
    float* hbuf   = alloc((size_t)MTOK * HID_);
    float* ybuf   = alloc((size_t)MTOK * HID_);   // also holds delta (8192 x 131)
    float* obuf   = alloc((size_t)MTOK * HID_);   // also holds upd   (8192 x 128)
    float* big    = alloc((size_t)MTOK * 1536);   // qkv (1152) / mlp (1536) / ln-gn (128)

    // --- pyramid ---
    {
        int t1 = S_ * CLAT * 24 * 32, t2 = S_ * CLAT * 12 * 16, t3 = S_ * CLAT * 6 * 8;
        pyramid_down<<<(t1 + 255) / 256, 256, 0, stream>>>(fmaps, pyr1, 48, 64);
        pyramid_down<<<(t2 + 255) / 256, 256, 0, stream>>>(pyr1, pyr2, 24, 32);
        pyramid_down<<<(t3 + 255) / 256, 256, 0, stream>>>(pyr2, pyr3, 12, 16);
    }
    // --- constant embeddings + state ---
    {
        int total = NTRK * EMB_ + S_ * EMB_;
        pos_time_embed<<<(total + 255) / 256, 256, 0, stream>>>(coords_init, posE, timE);
        int t = MTOK * CLAT;
        init_state<<<(t + 255) / 256, 256, 0, stream>>>(coords_init, feat_init, coords, ffeats);
    }

    for (int it = 0; it < iters; ++it) {
        // correlation features -> x[:,130:326)
        corr_sample<<<(S_ * NTRK) / 8, 256, 0, stream>>>(fmaps, pyr1, pyr2, pyr3, ffeats, coords, xbuf);
        // flow embedding GEMM -> x[:,0:130)
        flow_embed_gemm<<<MTOK, 256, 0, stream>>>(coords, W_flow, b_flow, xbuf);
        // assemble full input token
        assemble_x<<<(MTOK * EMB_ + 255) / 256, 256, 0, stream>>>(xbuf, ffeats, track_mask, vis_init, posE, timE);

        // h = x @ W_in + b_in
        launch_gemm(xbuf, W_in, b_in, hbuf, MTOK, EMB_, HID_, 0, stream);

        for (int l = 0; l < 3; ++l) {
            layernorm<<<MTOK / 8, 256, 0, stream>>>(hbuf, ln1_w + l * HID_, ln1_b + l * HID_, ybuf, HID_, HID_, 0);
            launch_gemm(ybuf, Wqkv + (size_t)l * HID_ * 3 * HID_, bqkv + l * 3 * HID_, big, MTOK, HID_, 3 * HID_, 0, stream);
            attention<<<(NTRK * NH_ * S_) / 256, 256, 0, stream>>>(big, obuf);
            launch_gemm(obuf, Wproj + (size_t)l * HID_ * HID_, bproj + l * HID_, hbuf, MTOK, HID_, HID_, 1, stream);
            layernorm<<<MTOK / 8, 256, 0, stream>>>(hbuf, ln2_w + l * HID_, ln2_b + l * HID_, ybuf, HID_, HID_, 0);
            launch_gemm(ybuf, Wm1 + (size_t)l * HID_ * 4 * HID_, bm1 + l * 4 * HID_, big, MTOK, HID_, 4 * HID_, 2, stream);
            launch_gemm(big, Wm2 + (size_t)l * 4 * HID_ * HID_, bm2 + l * HID_, hbuf, MTOK, 4 * HID_, HID_, 1, stream);
        }

        // delta = h @ W_out + b_out  (131 cols, stored in ybuf)
        launch_gemm(hbuf, W_out, b_out, ybuf, MTOK, HID_, CLAT + 3, 0, stream);
        // upd = gelu(LN_gn(delta[:,3:]) @ W_ffu + b_ffu)
        layernorm<<<MTOK / 8, 256, 0, stream>>>(ybuf, gn_w, gn_b, big, CLAT, CLAT + 3, 3);
        launch_gemm(big, W_ffu, b_ffu, obuf, MTOK, CLAT, CLAT, 2, stream);
        // ffeats += upd; coords += dc; write predictions
        update_state<<<(MTOK * CLAT) / 256, 256, 0, stream>>>(ybuf, obuf, ffeats, coords, out + (size_t)it * OUTSTEP);
    }

    vis_head<<<(S_ * NTRK + 255) / 256, 256, 0, stream>>>(ffeats, W_vis, b_vis, out + (size_t)iters * OUTSTEP);
}